// CrossLevelAttention_26938034880643
// MI455X (gfx1250) — compile-verified
//
#include <hip/hip_runtime.h>
#include <hip/hip_bf16.h>
#include <math.h>

typedef float v2f __attribute__((ext_vector_type(2)));
typedef float v8f __attribute__((ext_vector_type(8)));

#define NPIX 65536L   // 256*256 pixels per batch
#define HDIM 256
#define WDIM 256

// ---------------------------------------------------------------------------
// Generic 1x1 conv as GEMM: Y[b][co][n] = sum_ci W[co][ci] * X[b][ci][n]
// Uses V_WMMA_F32_16X16X4_F32 (full fp32). Each wave computes a 16(co)x16(n)
// tile; block = 8 waves covering 16co x 128n. grid = (N/128, Cout/16, B).
// A (16x4 f32): lane m=lane&15; VGPR j holds K = j + 2*(lane>=16)
// B (4x16 f32): lane n=lane&15; VGPR j holds K = j + 2*(lane>=16)
// C (16x16 f32): lane n=lane&15; VGPR j holds row m = j + 8*(lane>=16)
// ---------------------------------------------------------------------------
__global__ __launch_bounds__(256) void conv1x1_wmma(
    const float* __restrict__ X, const float* __restrict__ W,
    float* __restrict__ Y, int Cin, long xStrideB, long yStrideB, long wStrideB) {
  __shared__ float Wtile[16 * 260];  // stride 260 (260 % 64 = 4) avoids bank conflicts

  const int tid  = threadIdx.x;
  const int wave = tid >> 5;
  const int lane = tid & 31;
  const int half = lane >> 4;   // 0: lanes 0-15, 1: lanes 16-31
  const int l15  = lane & 15;

  const int  co_base = blockIdx.y * 16;
  const long n       = (long)blockIdx.x * 128 + wave * 16 + l15;
  const float* xb = X + (long)blockIdx.z * xStrideB;
  const float* wb = W + (long)blockIdx.z * wStrideB;

  // Stage the 16 x Cin weight tile into LDS (coalesced over k).
  for (int idx = tid; idx < 16 * Cin; idx += 256) {
    int r = idx / Cin, k = idx - r * Cin;
    Wtile[r * 260 + k] = wb[(long)(co_base + r) * Cin + k];
  }
  __syncthreads();

  v8f acc = {};
  for (int k = 0; k < Cin; k += 4) {
    v2f a, bm;
    a.x  = Wtile[l15 * 260 + k + 2 * half];
    a.y  = Wtile[l15 * 260 + k + 2 * half + 1];
    bm.x = xb[(long)(k + 2 * half) * NPIX + n];
    bm.y = xb[(long)(k + 2 * half + 1) * NPIX + n];
    acc = __builtin_amdgcn_wmma_f32_16x16x4_f32(
        /*neg_a=*/false, a, /*neg_b=*/false, bm,
        /*c_mod=*/(short)0, acc, /*reuse_a=*/false, /*reuse_b=*/false);
  }

  float* yb = Y + (long)blockIdx.z * yStrideB;
#pragma unroll
  for (int j = 0; j < 8; ++j)
    yb[(long)(co_base + j + 8 * half) * NPIX + n] = acc[j];
}

// ---------------------------------------------------------------------------
// Depthwise 3x3 pyramid with reflect padding. Channels split in 4 groups of
// C/4: group 0 = identity, groups 1..3 = dilated dw-conv (dilation = group).
// grid = (NPIX/256, C, B), block = 256.
// ---------------------------------------------------------------------------
__global__ __launch_bounds__(256) void pyramid_dw(
    const float* __restrict__ pre, float* __restrict__ post,
    const float* __restrict__ dw1, const float* __restrict__ dw2,
    const float* __restrict__ dw3, int C) {
  const int n = blockIdx.x * 256 + threadIdx.x;
  const int c = blockIdx.y, b = blockIdx.z;
  const int quarter = C >> 2;
  const int g = c / quarter;
  const float* src = pre + ((long)(b * C + c)) * NPIX;
  float* dst = post + ((long)(b * C + c)) * NPIX;
  if (g == 0) { dst[n] = src[n]; return; }
  const int d = g;
  const float* wgt = (g == 1 ? dw1 : (g == 2 ? dw2 : dw3)) + (c - g * quarter) * 9;
  const int y = n >> 8, xp = n & 255;
  float s = 0.f;
#pragma unroll
  for (int ky = 0; ky < 3; ++ky) {
    int yy = y + (ky - 1) * d;
    yy = yy < 0 ? -yy : (yy >= HDIM ? 2 * HDIM - 2 - yy : yy);
#pragma unroll
    for (int kx = 0; kx < 3; ++kx) {
      int xx = xp + (kx - 1) * d;
      xx = xx < 0 ? -xx : (xx >= WDIM ? 2 * WDIM - 2 - xx : xx);
      s += wgt[ky * 3 + kx] * src[yy * 256 + xx];
    }
  }
  dst[n] = s;
}

// ---------------------------------------------------------------------------
// Gram reduction over N: per (b, head) computes
//   Sh[c=0..7][d=0..15]  = sum_n qh[h*8+c][n] * k[h*16+d][n]
//   Sl[c=0..3][d=0..15]  = sum_n ql[h*4+c][n] * k[h*16+d][n]
//   sumsq of k / qh / ql rows (for the l2 norms)
// acc layout (floats): Sh@0 (1024) | Sl@1024 (512) | nk@1536 (128)
//                      | nqh@1664 (64) | nql@1728 (32)
// grid = (NPIX/512, 4 heads, 2 batches), block = 256.
// ---------------------------------------------------------------------------
#define RCHUNK 512
__global__ __launch_bounds__(256) void gram_reduce(
    const float* __restrict__ kv_post, const float* __restrict__ qh,
    const float* __restrict__ ql, float* __restrict__ acc) {
  __shared__ float lds[28][RCHUNK + 8];  // rows 0-15: k, 16-23: qh, 24-27: ql
  const int h = blockIdx.y, b = blockIdx.z;
  const long n0 = (long)blockIdx.x * RCHUNK;

  for (int idx = threadIdx.x; idx < 28 * RCHUNK; idx += 256) {
    int row = idx / RCHUNK, col = idx - row * RCHUNK;
    const float* src;
    if (row < 16)      src = kv_post + ((long)(b * 128 + h * 16 + row)) * NPIX;
    else if (row < 24) src = qh + ((long)(b * 32 + h * 8 + (row - 16))) * NPIX;
    else               src = ql + ((long)(b * 16 + h * 4 + (row - 24))) * NPIX;
    lds[row][col] = src[n0 + col];
  }
  __syncthreads();

  const int i = threadIdx.x;
  if (i < 220) {
    int ra, rb;
    if (i < 128)      { ra = 16 + (i >> 4); rb = i & 15; }            // Sh
    else if (i < 192) { int j = i - 128; ra = 24 + (j >> 4); rb = j & 15; } // Sl
    else if (i < 208) { ra = rb = i - 192; }                          // ||k||^2
    else if (i < 216) { ra = rb = 16 + (i - 208); }                   // ||qh||^2
    else              { ra = rb = 24 + (i - 216); }                   // ||ql||^2
    float s = 0.f;
    for (int p = 0; p < RCHUNK; ++p) s += lds[ra][p] * lds[rb][p];
    float* dst;
    if (i < 128)      dst = acc + ((b * 4 + h) * 8 + (i >> 4)) * 16 + (i & 15);
    else if (i < 192) dst = acc + 1024 + ((b * 4 + h) * 4 + ((i - 128) >> 4)) * 16 + ((i - 128) & 15);
    else if (i < 208) dst = acc + 1536 + b * 64 + h * 16 + (i - 192);
    else if (i < 216) dst = acc + 1664 + b * 32 + h * 8 + (i - 208);
    else              dst = acc + 1728 + b * 16 + h * 4 + (i - 216);
    atomicAdd(dst, s);
  }
}

// ---------------------------------------------------------------------------
// Finalize: normalize scores, softmax over d (16), fold with output 1x1 conv:
//   Wh_eff[b][o][h*16+d] = sum_c w_ph[o][h*8+c] * A_h[b][h][c][d]   (128x64)
//   Wl_eff[b][o][h*16+d] = sum_c w_pl[o][h*4+c] * A_l[b][h][c][d]   (64x64)
// grid = (2 batches), block = 256.
// ---------------------------------------------------------------------------
__global__ __launch_bounds__(256) void finalize_attn(
    const float* __restrict__ acc, const float* __restrict__ w_ph,
    const float* __restrict__ w_pl, const float* __restrict__ temperature,
    float* __restrict__ Wh_eff, float* __restrict__ Wl_eff) {
  __shared__ float Ah[32][16];
  __shared__ float Al[16][16];
  const int b = blockIdx.x, t = threadIdx.x;

  if (t < 48) {
    const bool isH = t < 32;
    const int r = isH ? t : t - 32;
    const int h = isH ? (r >> 3) : (r >> 2);
    const int c = isH ? (r & 7) : (r & 3);
    const float* S = isH ? acc + ((b * 4 + h) * 8 + c) * 16
                         : acc + 1024 + ((b * 4 + h) * 4 + c) * 16;
    float nq = isH ? acc[1664 + b * 32 + h * 8 + c]
                   : acc[1728 + b * 16 + h * 4 + c];
    nq = fmaxf(sqrtf(nq), 1e-12f);
    const float temp = temperature[h];
    float vals[16], mx = -1e30f;
#pragma unroll
    for (int d = 0; d < 16; ++d) {
      float nk = fmaxf(sqrtf(acc[1536 + b * 64 + h * 16 + d]), 1e-12f);
      vals[d] = S[d] / (nq * nk) * temp;
      mx = fmaxf(mx, vals[d]);
    }
    float sum = 0.f;
#pragma unroll
    for (int d = 0; d < 16; ++d) { vals[d] = expf(vals[d] - mx); sum += vals[d]; }
    const float inv = 1.f / sum;
#pragma unroll
    for (int d = 0; d < 16; ++d) {
      if (isH) Ah[h * 8 + c][d] = vals[d] * inv;
      else     Al[h * 4 + c][d] = vals[d] * inv;
    }
  }
  __syncthreads();

  for (int idx = t; idx < 128 * 64; idx += 256) {
    int o = idx >> 6, g = idx & 63, h = g >> 4, d = g & 15;
    float s = 0.f;
#pragma unroll
    for (int c = 0; c < 8; ++c) s += w_ph[o * 32 + h * 8 + c] * Ah[h * 8 + c][d];
    Wh_eff[b * 8192 + idx] = s;
  }
  for (int idx = t; idx < 64 * 64; idx += 256) {
    int o = idx >> 6, g = idx & 63, h = g >> 4, d = g & 15;
    float s = 0.f;
#pragma unroll
    for (int c = 0; c < 4; ++c) s += w_pl[o * 16 + h * 4 + c] * Al[h * 4 + c][d];
    Wl_eff[b * 4096 + idx] = s;
  }
}

__global__ void zero_kernel(float* p, int n) {
  int i = blockIdx.x * 256 + threadIdx.x;
  if (i < n) p[i] = 0.f;
}

// ---------------------------------------------------------------------------
extern "C" void kernel_launch(void* const* d_in, const int* in_sizes, int n_in,
                              void* d_out, int out_size, void* d_ws, size_t ws_size,
                              hipStream_t stream) {
  (void)in_sizes; (void)n_in; (void)out_size; (void)ws_size;
  const float* x      = (const float*)d_in[0];
  const float* x_h    = (const float*)d_in[1];
  const float* x_l    = (const float*)d_in[2];
  const float* w_kv   = (const float*)d_in[3];
  const float* kv_dw1 = (const float*)d_in[4];
  const float* kv_dw2 = (const float*)d_in[5];
  const float* kv_dw3 = (const float*)d_in[6];
  const float* w_qh   = (const float*)d_in[7];
  const float* qh_dw1 = (const float*)d_in[8];
  const float* qh_dw2 = (const float*)d_in[9];
  const float* qh_dw3 = (const float*)d_in[10];
  const float* w_ql   = (const float*)d_in[11];
  const float* ql_dw1 = (const float*)d_in[12];
  const float* ql_dw2 = (const float*)d_in[13];
  const float* ql_dw3 = (const float*)d_in[14];
  const float* w_ph   = (const float*)d_in[15];
  const float* w_pl   = (const float*)d_in[16];
  const float* temperature = (const float*)d_in[17];

  // Scratch layout (floats). Total ~46.2M floats (~185 MB).
  float* ws      = (float*)d_ws;
  float* kv_pre  = ws;                        // 2*128*NPIX
  float* kv_post = kv_pre  + 2 * 128 * NPIX;  // 2*128*NPIX (k: ch 0-63, v: ch 64-127)
  float* qh_pre  = kv_post + 2 * 128 * NPIX;  // 2*32*NPIX
  float* qh_post = qh_pre  + 2 * 32 * NPIX;   // 2*32*NPIX
  float* ql_pre  = qh_post + 2 * 32 * NPIX;   // 2*16*NPIX
  float* ql_post = ql_pre  + 2 * 16 * NPIX;   // 2*16*NPIX
  float* acc     = ql_post + 2 * 16 * NPIX;   // 1760
  float* Wh_eff  = acc + 1760;                // 2*128*64
  float* Wl_eff  = Wh_eff + 2 * 8192;         // 2*64*64

  dim3 blk(256);

  zero_kernel<<<dim3((1760 + 255) / 256), blk, 0, stream>>>(acc, 1760);

  // 1x1 convs (pre-pyramid) as WMMA GEMMs
  conv1x1_wmma<<<dim3(512, 8, 2), blk, 0, stream>>>(x,   w_kv, kv_pre, 256, 256 * NPIX, 128 * NPIX, 0);
  conv1x1_wmma<<<dim3(512, 2, 2), blk, 0, stream>>>(x_h, w_qh, qh_pre, 128, 128 * NPIX, 32 * NPIX, 0);
  conv1x1_wmma<<<dim3(512, 1, 2), blk, 0, stream>>>(x_l, w_ql, ql_pre, 64,  64 * NPIX,  16 * NPIX, 0);

  // Depthwise pyramids
  pyramid_dw<<<dim3(256, 128, 2), blk, 0, stream>>>(kv_pre, kv_post, kv_dw1, kv_dw2, kv_dw3, 128);
  pyramid_dw<<<dim3(256, 32, 2),  blk, 0, stream>>>(qh_pre, qh_post, qh_dw1, qh_dw2, qh_dw3, 32);
  pyramid_dw<<<dim3(256, 16, 2),  blk, 0, stream>>>(ql_pre, ql_post, ql_dw1, ql_dw2, ql_dw3, 16);

  // Gram matrices + norms over N
  gram_reduce<<<dim3(NPIX / RCHUNK, 4, 2), blk, 0, stream>>>(kv_post, qh_post, ql_post, acc);

  // Softmax + fold attention into effective output-projection weights
  finalize_attn<<<dim3(2), blk, 0, stream>>>(acc, w_ph, w_pl, temperature, Wh_eff, Wl_eff);

  // Output GEMMs directly into d_out: out1 = Wh_eff @ v, out2 = Wl_eff @ v
  float* out1 = (float*)d_out;
  float* out2 = out1 + 2 * 128 * NPIX;
  const float* v = kv_post + 64 * NPIX;  // v channels, per-batch stride 128*NPIX
  conv1x1_wmma<<<dim3(512, 8, 2), blk, 0, stream>>>(v, Wh_eff, out1, 64, 128 * NPIX, 128 * NPIX, 8192);
  conv1x1_wmma<<<dim3(512, 4, 2), blk, 0, stream>>>(v, Wl_eff, out2, 64, 128 * NPIX, 64 * NPIX, 4096);
}